// CasamentoMult_26431228739729
// MI455X (gfx1250) — compile-verified
//
#include <hip/hip_runtime.h>
#include <hip/hip_bf16.h>
#include <stdint.h>

// ---------------------------------------------------------------------------
// CasamentoMult on MI455X (gfx1250, wave32).
//
// Reference collapses to: sum over k in [0, K), K = N-2, of
//   1 + 0.5*(exp(-pi*(y1-y0)^2) + exp(-pi*(d1-d0)^2))
//     - 0.5*(exp(-pi*(d1-y1)^2)+exp(-pi*(d1-y0)^2)
//           +exp(-pi*(d0-y1)^2)+exp(-pi*(d0-y0)^2))
// with d0=d[k], d1=d[k+1], y0=y[k], y1=y[k+1].
//
// Memory: 32 MB @ 23.3 TB/s ~ 1.4us.  Compute: ~24M v_exp_f32 -- comparable.
// => stream tiles through LDS with gfx1250 async-to-LDS copies, double
//    buffered, so HBM fetch overlaps the exp-heavy VALU work.
// Final reduction of block partials uses V_WMMA_F32_16X16X4_F32 with an
// all-ones A matrix (layout-independent sum trick), one wave, deterministic.
// ---------------------------------------------------------------------------

#define BLOCK_T 256
#define NBLOCKS 512          // multiple of 64 (kernel-2 consumes 64/iter)
#define TILE    2048         // k-elements per tile
#define COPIES  9            // ceil((TILE+1)/BLOCK_T) -> uniform trip count

typedef __attribute__((ext_vector_type(2))) float v2f;
typedef __attribute__((ext_vector_type(8))) float v8f;

// gfx1250 async global->LDS copy (ASYNCcnt-tracked, fire-and-forget).
// lds_byte_addr: low 32 bits of the flat shared pointer == LDS byte offset.
__device__ __forceinline__ void async_copy_f32(unsigned lds_byte_addr,
                                               const float* gptr) {
  asm volatile("global_load_async_to_lds_b32 %0, %1, off"
               :: "v"(lds_byte_addr),
                  "v"((unsigned long long)(uintptr_t)gptr)
               : "memory");
}
__device__ __forceinline__ void wait_async_le18() {
  // 18 = async instructions issued per wave per stage (prefetch in flight)
  asm volatile("s_wait_asynccnt 0x12" ::: "memory");
}
__device__ __forceinline__ void wait_async_le0() {
  asm volatile("s_wait_asynccnt 0x0" ::: "memory");
}

// exp(-pi*t^2) = exp2(-pi*log2(e) * t^2) -> v_mul + v_exp_f32
__device__ __forceinline__ float gexp(float t) {
  const float C = -4.532360141827194f; // -pi * log2(e)
  return __builtin_amdgcn_exp2f(C * t * t);
}

// Stage elements base+j, j in [0, TILE], into LDS (clamped so every global
// read stays in-bounds; clamped slots are never consumed by compute).
// Uniform 2*COPIES = 18 async instructions per wave.
__device__ __forceinline__ void stage_tile(const float* __restrict__ d,
                                           const float* __restrict__ y,
                                           unsigned sdb, unsigned syb,
                                           long long base, long long K,
                                           int tid) {
#pragma unroll
  for (int r = 0; r < COPIES; ++r) {
    int j = tid + r * BLOCK_T;
    if (j > TILE) j = TILE;                 // redundant tail copies, same data
    long long e = base + (long long)j;
    if (e > K) e = K;                       // clamp: max valid index is K=N-2
    async_copy_f32(sdb + (unsigned)(j << 2), d + e);
    async_copy_f32(syb + (unsigned)(j << 2), y + e);
  }
}

__global__ __launch_bounds__(BLOCK_T) void casamento_main(
    const float* __restrict__ d, const float* __restrict__ y,
    float* __restrict__ partials, long long K) {
  __shared__ __align__(16) float sd[2][TILE + 4];
  __shared__ __align__(16) float sy[2][TILE + 4];
  __shared__ float red[BLOCK_T];

  const int tid = threadIdx.x;
  const long long tiles_total = (K + TILE - 1) / TILE;
  const long long stride = (long long)gridDim.x;

  unsigned sdb0 = (unsigned)(uintptr_t)&sd[0][0];
  unsigned sdb1 = (unsigned)(uintptr_t)&sd[1][0];
  unsigned syb0 = (unsigned)(uintptr_t)&sy[0][0];
  unsigned syb1 = (unsigned)(uintptr_t)&sy[1][0];

  float acc = 0.0f;
  long long t = (long long)blockIdx.x;
  int buf = 0;

  if (t < tiles_total)  // prologue: stage first tile into buffer 0
    stage_tile(d, y, sdb0, syb0, t * TILE, K, tid);

  for (; t < tiles_total; t += stride) {
    long long tn = t + stride;
    bool pref = (tn < tiles_total);
    if (pref) {  // prefetch next tile into the other buffer
      stage_tile(d, y, buf ? sdb0 : sdb1, buf ? syb0 : syb1, tn * TILE, K, tid);
      wait_async_le18();   // current tile's 18 copies done (in-order),
    } else {               // prefetch still in flight -> overlap with compute
      wait_async_le0();
    }
    __syncthreads();

    const float* __restrict__ pd = buf ? &sd[1][0] : &sd[0][0];
    const float* __restrict__ py = buf ? &sy[1][0] : &sy[0][0];
    long long base = t * TILE;
    int cnt = (K - base < (long long)TILE) ? (int)(K - base) : TILE;

    for (int i = tid; i < cnt; i += BLOCK_T) {
      float d0 = pd[i], d1 = pd[i + 1];
      float y0 = py[i], y1 = py[i + 1];
      float ey  = gexp(y1 - y0);
      float ed  = gexp(d1 - d0);
      float e11 = gexp(d1 - y1);
      float e10 = gexp(d1 - y0);
      float e01 = gexp(d0 - y1);
      float e00 = gexp(d0 - y0);
      acc += 1.0f + 0.5f * (ey + ed) - 0.5f * ((e11 + e10) + (e01 + e00));
    }
    __syncthreads();  // all waves done reading `buf` before it is re-staged
    buf ^= 1;
  }

  // Deterministic block reduction (no float atomics -> bitwise reproducible).
  red[tid] = acc;
  __syncthreads();
  for (int s = BLOCK_T / 2; s > 0; s >>= 1) {
    if (tid < s) red[tid] += red[tid + s];
    __syncthreads();
  }
  if (tid == 0) partials[blockIdx.x] = red[0];
}

// One wave32 reduces NBLOCKS partials with chained V_WMMA_F32_16X16X4_F32.
// A = ones(16x4) => D[i][j] = sum_k B[k][j] + C[i][j]; hence
// sum(D) = 16 * sum(all packed B values) regardless of A/B VGPR layout.
__global__ __launch_bounds__(32) void casamento_reduce(
    const float* __restrict__ partials, int P, float* __restrict__ out) {
  int lane = threadIdx.x;           // exactly one wave32, EXEC all ones
  v2f ones; ones.x = 1.0f; ones.y = 1.0f;
  v8f c = {};                       // C/D accumulator (8 VGPRs)

  for (int base = 0; base < P; base += 64) {
    v2f b;                          // pack 64 partials into B's 2 VGPRs
    b.x = partials[base + lane];
    b.y = partials[base + 32 + lane];
    c = __builtin_amdgcn_wmma_f32_16x16x4_f32(
        /*neg_a=*/false, ones, /*neg_b=*/false, b,
        /*c_mod=*/(short)0, c, /*reuse_a=*/false, /*reuse_b=*/false);
  }

  float s = ((c.s0 + c.s1) + (c.s2 + c.s3)) + ((c.s4 + c.s5) + (c.s6 + c.s7));
  for (int off = 16; off > 0; off >>= 1)
    s += __shfl_xor(s, off, 32);
  if (lane == 0) out[0] = s * (1.0f / 16.0f);
}

extern "C" void kernel_launch(void* const* d_in, const int* in_sizes, int n_in,
                              void* d_out, int out_size, void* d_ws,
                              size_t ws_size, hipStream_t stream) {
  const float* d = (const float*)d_in[0];
  const float* y = (const float*)d_in[1];
  long long N = (long long)in_sizes[0];
  long long K = N - 2;
  float* partials = (float*)d_ws;   // NBLOCKS * 4 bytes of scratch

  casamento_main<<<NBLOCKS, BLOCK_T, 0, stream>>>(d, y, partials, K);
  casamento_reduce<<<1, 32, 0, stream>>>(partials, NBLOCKS, (float*)d_out);
}